// GraphEncoder_65103114273323
// MI455X (gfx1250) — compile-verified
//
#include <hip/hip_runtime.h>
#include <hip/hip_bf16.h>

// ---------------------------------------------------------------------------
// GraphSAGE (2-layer, mean aggregation, d=128) for gfx1250.
//   layer: out = act( (segsum(x[src]->dst)/max(deg,1)) @ Wl^T + x @ Wr^T + b )
// Edge phase: bandwidth-bound scatter with f32 atomics (L2-resident accum).
// Dense phase: fp32 WMMA (v_wmma_f32_16x16x4_f32), one wave per 16x16 tile.
// ---------------------------------------------------------------------------

typedef __attribute__((ext_vector_type(2))) float v2f;
typedef __attribute__((ext_vector_type(8))) float v8f;

#define DFEAT 128

// ---------------- edge scatter: sum[dst] += feat[src], optional degree -----
template<bool COUNT_DEG>
__global__ __launch_bounds__(256) void sage_scatter(
    const float* __restrict__ feat,   // [N,128]
    const int*   __restrict__ src,    // [E]
    const int*   __restrict__ dst,    // [E]
    float* __restrict__ sum,          // [N,128] (zeroed)
    float* __restrict__ deg,          // [N]     (zeroed) -- used if COUNT_DEG
    int E)
{
    long long tid = (long long)blockIdx.x * blockDim.x + threadIdx.x;
    long long e   = tid >> 5;           // 32 lanes per edge
    int chunk     = (int)(tid & 31) * 4; // 4 floats per lane
    if (e >= E) return;
    int s = src[e];
    int d = dst[e];
    const float4 v = *(const float4*)(feat + (long long)s * DFEAT + chunk);
    float* o = sum + (long long)d * DFEAT + chunk;
    atomicAdd(o + 0, v.x);
    atomicAdd(o + 1, v.y);
    atomicAdd(o + 2, v.z);
    atomicAdd(o + 3, v.w);
    if (COUNT_DEG && chunk == 0) atomicAdd(deg + d, 1.0f);
}

// ---------------- deg -> 1/max(deg,1) in place -----------------------------
__global__ __launch_bounds__(256) void inv_deg_kernel(float* __restrict__ deg, int N)
{
    int i = blockIdx.x * blockDim.x + threadIdx.x;
    if (i < N) deg[i] = 1.0f / fmaxf(deg[i], 1.0f);
}

// ---------------- fused SAGE gemm: out = act(mean@Wl^T + x@Wr^T + b) -------
// One wave per 16(M) x 16(N-cols) tile; 8 waves/block -> 16 rows x 128 cols.
// K=128 consumed in steps of 4 via v_wmma_f32_16x16x4_f32, C chained.
template<bool RELU>
__global__ __launch_bounds__(256) void sage_gemm_wmma(
    const float* __restrict__ nsum,    // [N,128] neighbor sums
    const float* __restrict__ invdeg,  // [N]
    const float* __restrict__ xin,     // [N,128] root features
    const float* __restrict__ Wl,      // [128,128] row-major [out][in]
    const float* __restrict__ Wr,      // [128,128]
    const float* __restrict__ bias,    // [128]
    float* __restrict__ out)           // [N,128]
{
    const int wave  = threadIdx.x >> 5;        // 0..7 -> 16-col tile
    const int lane  = threadIdx.x & 31;
    const int r16   = lane & 15;               // row-in-tile / col-in-tile
    const int khalf = (lane >> 4) << 1;        // lanes>=16 carry K+2,K+3
    const int rowbase = blockIdx.x * 16;
    const int colbase = wave * 16;
    const int row = rowbase + r16;

    const float id = invdeg[row];

    const float* aL = nsum + (long long)row * DFEAT + khalf;
    const float* aR = xin  + (long long)row * DFEAT + khalf;
    const float* bL = Wl + (colbase + r16) * DFEAT + khalf;  // B[k][n] = Wl[n][k]
    const float* bR = Wr + (colbase + r16) * DFEAT + khalf;

    v8f acc = {};
#pragma unroll
    for (int k = 0; k < DFEAT; k += 4) {
        v2f am = *(const v2f*)(aL + k);
        am *= id;                                   // fold mean = sum * invdeg
        v2f bm = *(const v2f*)(bL + k);
        acc = __builtin_amdgcn_wmma_f32_16x16x4_f32(
            false, am, false, bm, (short)0, acc, false, false);
        v2f ax = *(const v2f*)(aR + k);
        v2f bx = *(const v2f*)(bR + k);
        acc = __builtin_amdgcn_wmma_f32_16x16x4_f32(
            false, ax, false, bx, (short)0, acc, false, false);
    }

    // D layout: VGPR r -> M=r (lanes 0-15) / M=r+8 (lanes 16-31), N=lane&15.
    const float bv = bias[colbase + r16];
    float* op = out + (long long)(rowbase + (lane >> 4) * 8) * DFEAT + colbase + r16;
#pragma unroll
    for (int r = 0; r < 8; ++r) {
        float v = acc[r] + bv;
        if (RELU) v = fmaxf(v, 0.0f);
        op[(long long)r * DFEAT] = v;
    }
}

// ---------------- scalar tail for rows not covered by 16-row tiles ---------
template<bool RELU>
__global__ __launch_bounds__(128) void sage_gemm_tail(
    const float* __restrict__ nsum, const float* __restrict__ invdeg,
    const float* __restrict__ xin,  const float* __restrict__ Wl,
    const float* __restrict__ Wr,   const float* __restrict__ bias,
    float* __restrict__ out, int row0, int N)
{
    int row = row0 + blockIdx.x;
    if (row >= N) return;
    int j = threadIdx.x;               // output feature
    float id = invdeg[row];
    const float* a = nsum + (long long)row * DFEAT;
    const float* x = xin  + (long long)row * DFEAT;
    const float* wl = Wl + j * DFEAT;
    const float* wr = Wr + j * DFEAT;
    float acc = bias[j];
    for (int k = 0; k < DFEAT; ++k)
        acc += a[k] * id * wl[k] + x[k] * wr[k];
    if (RELU) acc = fmaxf(acc, 0.0f);
    out[(long long)row * DFEAT + j] = acc;
}

extern "C" void kernel_launch(void* const* d_in, const int* in_sizes, int n_in,
                              void* d_out, int out_size, void* d_ws, size_t ws_size,
                              hipStream_t stream) {
    const float* x   = (const float*)d_in[0];
    const int*   ei  = (const int*)  d_in[1];
    const float* W1l = (const float*)d_in[2];
    const float* W1r = (const float*)d_in[3];
    const float* b1  = (const float*)d_in[4];
    const float* W2l = (const float*)d_in[5];
    const float* W2r = (const float*)d_in[6];
    const float* b2  = (const float*)d_in[7];

    const int N = in_sizes[0] / DFEAT;
    const int E = in_sizes[1] / 2;
    const int* src = ei;
    const int* dst = ei + E;

    // workspace layout: deg[N] | sum[N*128] | h[N*128]
    float* deg = (float*)d_ws;
    size_t degPad = ((size_t)N + 255) & ~(size_t)255;
    float* sum = deg + degPad;
    float* h   = sum + (size_t)N * DFEAT;
    float* outp = (float*)d_out;

    const int scatterBlocks = (int)(((long long)E * 32 + 255) / 256);
    const int tileBlocks    = N / 16;
    const int remRows       = N - tileBlocks * 16;

    // ---- layer 1 ----
    hipMemsetAsync(deg, 0, (size_t)N * sizeof(float), stream);
    hipMemsetAsync(sum, 0, (size_t)N * DFEAT * sizeof(float), stream);
    sage_scatter<true><<<scatterBlocks, 256, 0, stream>>>(x, src, dst, sum, deg, E);
    inv_deg_kernel<<<(N + 255) / 256, 256, 0, stream>>>(deg, N);
    sage_gemm_wmma<true><<<tileBlocks, 256, 0, stream>>>(sum, deg, x, W1l, W1r, b1, h);
    if (remRows > 0)
        sage_gemm_tail<true><<<remRows, 128, 0, stream>>>(sum, deg, x, W1l, W1r, b1, h,
                                                          tileBlocks * 16, N);

    // ---- layer 2 ----
    hipMemsetAsync(sum, 0, (size_t)N * DFEAT * sizeof(float), stream);
    sage_scatter<false><<<scatterBlocks, 256, 0, stream>>>(h, src, dst, sum, deg, E);
    sage_gemm_wmma<false><<<tileBlocks, 256, 0, stream>>>(sum, deg, h, W2l, W2r, b2, outp);
    if (remRows > 0)
        sage_gemm_tail<false><<<remRows, 128, 0, stream>>>(sum, deg, h, W2l, W2r, b2, outp,
                                                           tileBlocks * 16, N);
}